// GravityDisplacement_32856499815185
// MI455X (gfx1250) — compile-verified
//
#include <hip/hip_runtime.h>
#include <math.h>

typedef __attribute__((ext_vector_type(2))) float v2f;
typedef __attribute__((ext_vector_type(8))) float v8f;

#define LDIM  1024
#define DDIM  256
#define HDIM  256
#define H2DIM 128

constexpr float SURF      = 103.0f;
constexpr float SPACING   = SURF / 31.0f;          // N_ROW-1 = 31
constexpr float SMINC     = -SURF * 0.5f;
constexpr float SMAXC     =  SURF * 0.5f;
constexpr float DANGER    = SPACING * 0.5f;
constexpr float SIGMA     = SPACING * 0.5f;
constexpr float DSTEP     = SPACING * 0.1f;
constexpr float MAX_DSTEP = SPACING * 0.25f;
constexpr float MAX_TOT   = SPACING * 0.5f;
constexpr float MAX_DISP  = 3.0f;
constexpr float MIN_DISP  = 0.5f;
constexpr float REPULSE   = 0.5f;

__device__ __forceinline__ float gelu_exact(float x) {
    return 0.5f * x * (1.0f + erff(x * 0.70710678118654752440f));
}
__device__ __forceinline__ float softplus_f(float x) {
    return (x > 20.0f) ? x : log1pf(expf(x));
}
__device__ __forceinline__ float clampf(float x, float lo, float hi) {
    return fminf(fmaxf(x, lo), hi);
}

// ---------------------------------------------------------------------------
// Kernel 1: fused MLP  latents[16 rows] -> errors[16]
// GEMM1(256x256) + bias + LayerNorm + GELU + GEMM2(256x128) + bias + GELU
// + matvec(128->1) + softplus.  All GEMMs on the fp32 WMMA pipe.
// Block = 256 threads = 8 waves; one block per 16-row tile.
// ---------------------------------------------------------------------------
__global__ __launch_bounds__(256) void mlp_errors_kernel(
    const float* __restrict__ latents,
    const float* __restrict__ w1, const float* __restrict__ b1,
    const float* __restrict__ ln_g, const float* __restrict__ ln_b,
    const float* __restrict__ w2, const float* __restrict__ b2,
    const float* __restrict__ w3, const float* __restrict__ b3,
    float* __restrict__ errors)
{
    __shared__ float h1[16 * HDIM];    // 16 KB
    __shared__ float h2[16 * H2DIM];   //  8 KB
    __shared__ float mu[16];
    __shared__ float rstd[16];

    const int t    = threadIdx.x;
    const int lane = t & 31;
    const int wave = t >> 5;
    const long rowBase = (long)blockIdx.x * 16;

    // fp32 16x16x4 WMMA fragment coordinates (ISA 7.12.2)
    const int m    = lane & 15;              // A row / also C col index source
    const int kh   = (lane >> 4) << 1;       // K offset: 0 (lanes 0-15) or 2
    const int n    = lane & 15;              // B/C/D column
    const int mrow = (lane >> 4) << 3;       // C/D row base: 0 or 8

    // ---- GEMM1: [16 x 256] @ w1[256 x 256] + b1 -> h1 (LDS) ----
    const float* arow = latents + (rowBase + m) * DDIM + kh;
    for (int rep = 0; rep < 2; ++rep) {
        const int n0 = (wave + rep * 8) * 16;     // 16 N-tiles, 2 per wave
        v8f acc = {};
        for (int k = 0; k < DDIM; k += 4) {
            v2f a, bfr;
            a.x = arow[k];
            a.y = arow[k + 1];
            const float* bp = w1 + (k + kh) * HDIM + n0 + n;
            bfr.x = bp[0];
            bfr.y = bp[HDIM];
            acc = __builtin_amdgcn_wmma_f32_16x16x4_f32(
                false, a, false, bfr, (short)0, acc, false, false);
        }
        #pragma unroll
        for (int r = 0; r < 8; ++r)
            h1[(mrow + r) * HDIM + n0 + n] = acc[r] + b1[n0 + n];
    }
    __syncthreads();

    // ---- LayerNorm stats per row ----
    if (t < 16) {
        float s = 0.0f;
        for (int j = 0; j < HDIM; ++j) s += h1[t * HDIM + j];
        const float mean = s / (float)HDIM;
        float v = 0.0f;
        for (int j = 0; j < HDIM; ++j) {
            const float d = h1[t * HDIM + j] - mean;
            v += d * d;
        }
        mu[t]   = mean;
        rstd[t] = rsqrtf(v / (float)HDIM + 1e-5f);
    }
    __syncthreads();

    // ---- LN scale/shift + exact GELU, elementwise ----
    for (int e = t; e < 16 * HDIM; e += 256) {
        const int r = e >> 8;            // / HDIM
        const int c = e & (HDIM - 1);
        const float x = (h1[e] - mu[r]) * rstd[r] * ln_g[c] + ln_b[c];
        h1[e] = gelu_exact(x);
    }
    __syncthreads();

    // ---- GEMM2: h1[16 x 256] @ w2[256 x 128] + b2, GELU -> h2 (LDS) ----
    {
        const int n0 = wave * 16;                 // 8 N-tiles, 1 per wave
        v8f acc = {};
        for (int k = 0; k < HDIM; k += 4) {
            v2f a, bfr;
            a.x = h1[m * HDIM + k + kh];
            a.y = h1[m * HDIM + k + kh + 1];
            const float* bp = w2 + (k + kh) * H2DIM + n0 + n;
            bfr.x = bp[0];
            bfr.y = bp[H2DIM];
            acc = __builtin_amdgcn_wmma_f32_16x16x4_f32(
                false, a, false, bfr, (short)0, acc, false, false);
        }
        #pragma unroll
        for (int r = 0; r < 8; ++r)
            h2[(mrow + r) * H2DIM + n0 + n] = gelu_exact(acc[r] + b2[n0 + n]);
    }
    __syncthreads();

    // ---- GEMM3 matvec: h2[16 x 128] @ w3[128] + b3, softplus ----
    if (t < 16) {
        float s = b3[0];
        for (int j = 0; j < H2DIM; ++j) s += h2[t * H2DIM + j] * w3[j];
        errors[rowBase + t] = softplus_f(s);
    }
}

// ---------------------------------------------------------------------------
// Kernel 2: robust normalization of errors over L per batch.
// ---------------------------------------------------------------------------
__global__ __launch_bounds__(256) void robust_norm_kernel(
    const float* __restrict__ errors, float* __restrict__ en)
{
    __shared__ float rmin[256];
    __shared__ float rmax[256];
    const int b = blockIdx.x;
    const int t = threadIdx.x;
    const float* e = errors + (size_t)b * LDIM;

    float el[4];
    float mn = 1e30f, mx = -1e30f;
    #pragma unroll
    for (int q = 0; q < 4; ++q) {
        el[q] = log1pf(e[t + q * 256]);
        mn = fminf(mn, el[q]);
        mx = fmaxf(mx, el[q]);
    }
    rmin[t] = mn;
    rmax[t] = mx;
    __syncthreads();
    for (int s = 128; s > 0; s >>= 1) {
        if (t < s) {
            rmin[t] = fminf(rmin[t], rmin[t + s]);
            rmax[t] = fmaxf(rmax[t], rmax[t + s]);
        }
        __syncthreads();
    }
    const float lo  = rmin[0];
    const float inv = 1.0f / fmaxf(rmax[0] - lo, 1e-6f);
    #pragma unroll
    for (int q = 0; q < 4; ++q)
        en[(size_t)b * LDIM + t + q * 256] = (el[q] - lo) * inv;
}

// ---------------------------------------------------------------------------
// Kernel 3: forces + displacement + 3 density-spread iterations.
// One block per batch; positions/errors resident in LDS; in-block barriers
// replace inter-kernel global syncs for the Jacobi iterations.
// ---------------------------------------------------------------------------
__global__ __launch_bounds__(256) void dynamics_kernel(
    const float* __restrict__ positions,
    const float* __restrict__ en_g,
    float* __restrict__ out)
{
    __shared__ float px[LDIM], py[LDIM];
    __shared__ float sx[LDIM], sy[LDIM];
    __shared__ float en[LDIM], anom[LDIM];
    __shared__ float red[256];

    const int b = blockIdx.x;
    const int t = threadIdx.x;
    const float* pos = positions + (size_t)b * LDIM * 2;

    // load positions + errors, accumulate error sum
    float esum = 0.0f;
    for (int k = t; k < LDIM; k += 256) {
        px[k] = pos[2 * k];
        py[k] = pos[2 * k + 1];
        const float e = en_g[(size_t)b * LDIM + k];
        en[k] = e;
        esum += e;
    }
    red[t] = esum;
    __syncthreads();
    for (int s = 128; s > 0; s >>= 1) {
        if (t < s) red[t] += red[t + s];
        __syncthreads();
    }
    const float emean = red[0] / (float)LDIM;
    __syncthreads();
    for (int k = t; k < LDIM; k += 256) anom[k] = en[k] - emean;
    __syncthreads();

    // ---- gravity + repulsion forces, 4 points per thread ----
    float fx[4], fy[4], mag[4];
    float msum = 0.0f;
    #pragma unroll
    for (int q = 0; q < 4; ++q) {
        const int i = t + q * 256;
        const float xi = px[i], yi = py[i];
        float gx = 0.0f, gy = 0.0f;
        for (int j = 0; j < LDIM; ++j) {
            if (j == i) continue;
            const float dx = px[j] - xi;
            const float dy = py[j] - yi;
            const float d2 = dx * dx + dy * dy;
            const float dist = fmaxf(sqrtf(d2 + 1e-12f), 1e-6f);
            const float invd = 1.0f / dist;
            const float viol = fmaxf(DANGER - dist, 0.0f);
            const float coef = anom[j] * invd * invd
                             - REPULSE * expm1f(viol / DANGER);
            gx += dx * invd * coef;
            gy += dy * invd * coef;
        }
        fx[q] = gx;
        fy[q] = gy;
        mag[q] = sqrtf(gx * gx + gy * gy + 1e-16f);
        msum += mag[q];
    }
    red[t] = msum;
    __syncthreads();
    for (int s = 128; s > 0; s >>= 1) {
        if (t < s) red[t] += red[t + s];
        __syncthreads();
    }
    const float magmean = red[0] / (float)LDIM;
    __syncthreads();

    // ---- bounded displacement; write phase-1 positions (all reads done) ----
    #pragma unroll
    for (int q = 0; q < 4; ++q) {
        const int i = t + q * 256;
        const float mg   = mag[q];
        const float invm = 1.0f / (mg + 1e-8f);
        float rel = mg / (magmean + 1e-8f);
        rel = clampf(rel, 0.0f, 2.0f);
        const float dm = MIN_DISP + rel * 0.5f * (MAX_DISP - MIN_DISP);
        const float nx = clampf(px[i] + fx[q] * invm * dm, SMINC, SMAXC);
        const float ny = clampf(py[i] + fy[q] * invm * dm, SMINC, SMAXC);
        px[i] = nx; py[i] = ny;
        sx[i] = nx; sy[i] = ny;   // pos_start for density phase
    }
    __syncthreads();

    // ---- 3 Jacobi iterations of error-aware density spreading ----
    const float wcoef = 1.0f / (2.0f * SIGMA * SIGMA);
    const float gcoef = -1.0f / (SIGMA * SIGMA);
    for (int it = 0; it < 3; ++it) {
        float nxq[4], nyq[4];
        #pragma unroll
        for (int q = 0; q < 4; ++q) {
            const int i = t + q * 256;
            const float xi = px[i], yi = py[i];
            float ax = 0.0f, ay = 0.0f;
            for (int j = 0; j < LDIM; ++j) {
                if (j == i) continue;
                const float dx = xi - px[j];
                const float dy = yi - py[j];
                const float w = expf(-(dx * dx + dy * dy) * wcoef);
                ax += dx * w;
                ay += dy * w;
            }
            const float gx = ax * gcoef;
            const float gy = ay * gcoef;
            const float strength = 1.0f - en[i];
            float stx = -gx * DSTEP * strength;
            float sty = -gy * DSTEP * strength;
            const float sm = sqrtf(stx * stx + sty * sty + 1e-16f);
            const float sc = fminf(1.0f, MAX_DSTEP / (sm + 1e-8f));
            float pxn = xi + stx * sc;
            float pyn = yi + sty * sc;
            const float tx = pxn - sx[i];
            const float ty = pyn - sy[i];
            const float tm = sqrtf(tx * tx + ty * ty + 1e-16f);
            const float tc = fminf(1.0f, MAX_TOT / (tm + 1e-8f));
            nxq[q] = clampf(sx[i] + tx * tc, SMINC, SMAXC);
            nyq[q] = clampf(sy[i] + ty * tc, SMINC, SMAXC);
        }
        __syncthreads();   // all reads of px/py complete
        #pragma unroll
        for (int q = 0; q < 4; ++q) {
            const int i = t + q * 256;
            px[i] = nxq[q];
            py[i] = nyq[q];
        }
        __syncthreads();   // all writes visible
    }

    // ---- output ----
    float* o = out + (size_t)b * LDIM * 2;
    for (int k = t; k < LDIM; k += 256) {
        o[2 * k]     = px[k];
        o[2 * k + 1] = py[k];
    }
}

// ---------------------------------------------------------------------------
extern "C" void kernel_launch(void* const* d_in, const int* in_sizes, int n_in,
                              void* d_out, int out_size, void* d_ws, size_t ws_size,
                              hipStream_t stream) {
    const float* latents   = (const float*)d_in[0];
    const float* positions = (const float*)d_in[1];
    const float* w1   = (const float*)d_in[2];
    const float* b1   = (const float*)d_in[3];
    const float* ln_g = (const float*)d_in[4];
    const float* ln_b = (const float*)d_in[5];
    const float* w2   = (const float*)d_in[6];
    const float* b2   = (const float*)d_in[7];
    const float* w3   = (const float*)d_in[8];
    const float* b3   = (const float*)d_in[9];

    const int B = in_sizes[0] / (LDIM * DDIM);

    float* errors = (float*)d_ws;               // [B*L]
    float* en     = errors + (size_t)B * LDIM;  // [B*L]

    mlp_errors_kernel<<<B * LDIM / 16, 256, 0, stream>>>(
        latents, w1, b1, ln_g, ln_b, w2, b2, w3, b3, errors);
    robust_norm_kernel<<<B, 256, 0, stream>>>(errors, en);
    dynamics_kernel<<<B, 256, 0, stream>>>(positions, en, (float*)d_out);
}